// HGWaveNet_9019431321780
// MI455X (gfx1250) — compile-verified
//
#include <hip/hip_runtime.h>

typedef float v2f __attribute__((ext_vector_type(2)));
typedef float v8f __attribute__((ext_vector_type(8)));

#define D 128

// ---------------- zero fill (float4 granularity) ----------------
__global__ __launch_bounds__(256) void hg_zero(float4* p, int n4) {
    int i = blockIdx.x * 256 + threadIdx.x;
    if (i < n4) p[i] = make_float4(0.f, 0.f, 0.f, 0.f);
}

// ---------------- degree counting ----------------
__global__ __launch_bounds__(256) void hg_degrees(const int* __restrict__ src,
                                                  const int* __restrict__ dst,
                                                  float* __restrict__ deg_out,
                                                  float* __restrict__ deg_in,
                                                  int n_edges) {
    int e = blockIdx.x * 256 + threadIdx.x;
    if (e < n_edges) {
        atomicAdd(deg_out + src[e], 1.0f);
        atomicAdd(deg_in  + dst[e], 1.0f);
    }
}

// ---------------- deg -> rsqrt(max(deg,1)) in place ----------------
__global__ __launch_bounds__(256) void hg_norms(float* deg_out, float* deg_in, int n) {
    int i = blockIdx.x * 256 + threadIdx.x;
    if (i < n) {
        deg_out[i] = rsqrtf(fmaxf(deg_out[i], 1.0f));
        deg_in[i]  = rsqrtf(fmaxf(deg_in[i],  1.0f));
    }
}

// ---------------- fold weights ----------------
// W0p[(i>>1)*256 + o*2 + (i&1)] = tcn_w[o][i][0]
// Wcp[  same layout            ] = sum_j gc_w[i][j] * tcn_w[o][j][1]
// bias_c[o] = tcn_b[o] + sum_j gc_b[j] * tcn_w[o][j][1]
__global__ __launch_bounds__(128) void hg_prep_weights(const float* __restrict__ gc_w,
                                                       const float* __restrict__ gc_b,
                                                       const float* __restrict__ tcn_w,
                                                       const float* __restrict__ tcn_b,
                                                       float* __restrict__ W0p,
                                                       float* __restrict__ Wcp,
                                                       float* __restrict__ bias_c) {
    int i = blockIdx.x;    // input channel
    int o = threadIdx.x;   // output channel
    float w0 = tcn_w[o * (D * 3) + i * 3 + 0];
    float acc = 0.f;
    for (int j = 0; j < D; ++j)
        acc += gc_w[i * D + j] * tcn_w[o * (D * 3) + j * 3 + 1];
    int idx = (i >> 1) * (2 * D) + o * 2 + (i & 1);
    W0p[idx] = w0;
    Wcp[idx] = acc;
    if (i == 0) {
        float bb = tcn_b[o];
        for (int j = 0; j < D; ++j)
            bb += gc_b[j] * tcn_w[o * (D * 3) + j * 3 + 1];
        bias_c[o] = bb;
    }
}

// ---------------- edge gather/scale/scatter: agg[dst] += x[src]*norm_out[src] ----------------
// one wave32 per edge, 4 floats per lane
__global__ __launch_bounds__(256) void hg_edge_scatter(const float* __restrict__ x,
                                                       const int* __restrict__ src,
                                                       const int* __restrict__ dst,
                                                       const float* __restrict__ norm_out,
                                                       float* __restrict__ agg,
                                                       int n_edges) {
    int e = blockIdx.x * 8 + (threadIdx.x >> 5);
    if (e >= n_edges) return;
    int lane = threadIdx.x & 31;
    int s = src[e];
    int d = dst[e];
    float no = norm_out[s];
    float4 v = *(const float4*)(x + (size_t)s * D + lane * 4);
    float* base = agg + (size_t)d * D + lane * 4;
    atomicAdd(base + 0, v.x * no);
    atomicAdd(base + 1, v.y * no);
    atomicAdd(base + 2, v.z * no);
    atomicAdd(base + 3, v.w * no);
}

// ---------------- fused output GEMM with fp32 WMMA ----------------
// out[m][o] = bias_c[o] + sum_k hist2[m][k]*W0p(k,o) + norm_in[m]*sum_k agg[m][k]*Wcp(k,o)
// agg aliases out (d_out). Each wave owns a full 16-row output stripe (all 8
// column tiles -> 8 v8f accumulators), so each A fragment feeds 8 WMMAs and
// no wave touches another wave's rows (no barrier needed despite aliasing).
__global__ __launch_bounds__(256) void hg_gemm_out(const float* __restrict__ hist2,
                                                   float* __restrict__ agg_out,   // agg in, out overwritten
                                                   const float* __restrict__ norm_in,
                                                   const float* __restrict__ W0p,
                                                   const float* __restrict__ Wcp,
                                                   const float* __restrict__ bias_c,
                                                   int n_nodes) {
    const int wave = threadIdx.x >> 5;
    const int lane = threadIdx.x & 31;
    const int lo   = lane & 15;
    const int hi   = lane >> 4;
    const int tile = blockIdx.x * 8 + wave;   // 16-row tile index

    int mA = tile * 16 + lo;                  // A-row for this lane (feeds only output row mA)
    if (mA >= n_nodes) mA = n_nodes - 1;      // clamp: garbage rows are never stored
    const float nin = norm_in[mA];

    const float* arow_h = hist2   + (size_t)mA * D;
    const float* arow_g = agg_out + (size_t)mA * D;

    v8f c[8];
    #pragma unroll
    for (int ot = 0; ot < 8; ++ot) {
        float b = bias_c[ot * 16 + lo];
        #pragma unroll
        for (int r = 0; r < 8; ++r) c[ot][r] = b;
    }

    for (int kb = 0; kb < D / 4; ++kb) {
        const int k0 = kb * 4 + hi * 2;                    // even
        v2f a_h = *(const v2f*)(arow_h + k0);
        v2f a_g = *(const v2f*)(arow_g + k0);
        a_g[0] *= nin;
        a_g[1] *= nin;
        const float* brow0 = W0p + (kb * 2 + hi) * (2 * D) + lo * 2;
        const float* browc = Wcp + (kb * 2 + hi) * (2 * D) + lo * 2;
        #pragma unroll
        for (int ot = 0; ot < 8; ++ot) {
            v2f b0 = *(const v2f*)(brow0 + ot * 32);
            v2f bc = *(const v2f*)(browc + ot * 32);
            c[ot] = __builtin_amdgcn_wmma_f32_16x16x4_f32(false, a_h, false, b0,
                                                          (short)0, c[ot], false, false);
            c[ot] = __builtin_amdgcn_wmma_f32_16x16x4_f32(false, a_g, false, bc,
                                                          (short)0, c[ot], false, false);
        }
    }

    float* outrow = agg_out + (size_t)(tile * 16 + hi * 8) * D + lo;
    if (tile * 16 + 16 <= n_nodes) {
        // fast path: whole tile in range, no exec-mask juggling
        #pragma unroll
        for (int ot = 0; ot < 8; ++ot)
            #pragma unroll
            for (int r = 0; r < 8; ++r)
                outrow[(size_t)r * D + ot * 16] = c[ot][r];
    } else {
        #pragma unroll
        for (int ot = 0; ot < 8; ++ot)
            #pragma unroll
            for (int r = 0; r < 8; ++r) {
                int m = tile * 16 + r + hi * 8;
                if (m < n_nodes) agg_out[(size_t)m * D + ot * 16 + lo] = c[ot][r];
            }
    }
}

extern "C" void kernel_launch(void* const* d_in, const int* in_sizes, int n_in,
                              void* d_out, int out_size, void* d_ws, size_t ws_size,
                              hipStream_t stream) {
    const float* x      = (const float*)d_in[0];   // node_embeddings [N,128]
    const float* gc_w   = (const float*)d_in[1];   // [128,128]
    const float* gc_b   = (const float*)d_in[2];   // [128]
    const float* tcn_w  = (const float*)d_in[3];   // [128,128,3]
    const float* tcn_b  = (const float*)d_in[4];   // [128]
    const float* hist2  = (const float*)d_in[7];   // [N,128]
    const int*   src    = (const int*)d_in[8];
    const int*   dst    = (const int*)d_in[9];

    const int N = in_sizes[0] / D;
    const int E = in_sizes[8];

    float* ws      = (float*)d_ws;
    float* deg_out = ws;                    // N floats -> becomes norm_out
    float* deg_in  = ws + N;                // N floats -> becomes norm_in
    float* W0p     = ws + 2 * N;            // 128*128 packed
    float* Wcp     = W0p + D * D;           // 128*128 packed
    float* bias_c  = Wcp + D * D;           // 128

    float* agg = (float*)d_out;             // agg aliases output buffer [N,128]

    // 1) zero degree arrays and agg
    {
        int n4 = (2 * N + 3) / 4;
        hg_zero<<<(n4 + 255) / 256, 256, 0, stream>>>((float4*)deg_out, n4);
        int a4 = (N * D) / 4;
        hg_zero<<<(a4 + 255) / 256, 256, 0, stream>>>((float4*)agg, a4);
    }
    // 2) degrees
    hg_degrees<<<(E + 255) / 256, 256, 0, stream>>>(src, dst, deg_out, deg_in, E);
    // 3) rsqrt norms (in place)
    hg_norms<<<(N + 255) / 256, 256, 0, stream>>>(deg_out, deg_in, N);
    // 4) fold weights
    hg_prep_weights<<<D, D, 0, stream>>>(gc_w, gc_b, tcn_w, tcn_b, W0p, Wcp, bias_c);
    // 5) edge gather/scatter
    hg_edge_scatter<<<(E + 7) / 8, 256, 0, stream>>>(x, src, dst, deg_out, agg, E);
    // 6) fused WMMA GEMM -> out (one wave per 16-row stripe, 8 stripes/block)
    hg_gemm_out<<<(N + 127) / 128, 256, 0, stream>>>(hist2, agg, deg_in, W0p, Wcp, bias_c, N);
}